// PyTorchVoxelizer_57062935495127
// MI455X (gfx1250) — compile-verified
//
#include <hip/hip_runtime.h>
#include <hip/hip_bf16.h>

typedef float v2f __attribute__((ext_vector_type(2)));
typedef float v8f __attribute__((ext_vector_type(8)));
typedef int   v4i __attribute__((ext_vector_type(4)));

#define NG    8192
#define NF    32
#define GX    200
#define GY    200
#define GZ    16
#define VOXSZ 0.5f
#define VMINX (-50.f)
#define VMINY (-50.f)
#define VMINZ (-4.f)
#define VMAXX (50.f)
#define VMAXY (50.f)
#define VMAXZ (4.f)
#define EPSV  1e-6f
#define SENT  0xFFFFu
#define BATCH 512

// 64-byte per-gaussian record: mean, opacity, inv-cov (off-diag pre-doubled), bbox
struct __align__(16) GB {
    float mx, my, mz, op;
    float i00, i01, i02, i11;   // i01,i02 pre-doubled
    float i12, i22;             // i12 pre-doubled
    int   lx, ly;
    int   lz, hx, hy, hz;
};
static_assert(sizeof(GB) == 64, "GB must be 64B");

struct PK { int lo, hi; };      // bbox packed 10 bits/coord, biased +1

#if __has_builtin(__builtin_amdgcn_global_load_async_to_lds_b128) && \
    __has_builtin(__builtin_amdgcn_s_wait_asynccnt)
#define USE_ASYNC 1
typedef __attribute__((address_space(1))) v4i GlobV4;
typedef __attribute__((address_space(3))) v4i LdsV4;
#else
#define USE_ASYNC 0
#endif

__device__ __forceinline__ void copy16_g2l(const float* __restrict__ gsrc,
                                           float* __restrict__ ldst) {
#if USE_ASYNC
    __builtin_amdgcn_global_load_async_to_lds_b128(
        (GlobV4*)gsrc, (LdsV4*)ldst, 0, 0);
#else
    *(uint4*)ldst = *(const uint4*)gsrc;
#endif
}

// ---------------------------------------------------------------------------
// Kernel 1: per-gaussian preprocessing (cov^-1, bbox ints, validity)
// ---------------------------------------------------------------------------
__global__ void prep_kernel(const float* __restrict__ means,
                            const float* __restrict__ ops,
                            const float* __restrict__ cov,
                            GB* __restrict__ gb,
                            PK* __restrict__ pk)
{
    int i = blockIdx.x * blockDim.x + threadIdx.x;
    if (i >= NG) return;

    float c00 = cov[i*9+0], c01 = cov[i*9+1], c02 = cov[i*9+2];
    float c11 = cov[i*9+4], c12 = cov[i*9+5], c22 = cov[i*9+8];
    float mx  = means[i*3+0], my = means[i*3+1], mz = means[i*3+2];
    float op  = ops[i];

    float sx = sqrtf(c00), sy = sqrtf(c11), sz = sqrtf(c22);
    float lox = mx - 3.f*sx, loy = my - 3.f*sy, loz = mz - 3.f*sz;
    float upx = mx + 3.f*sx, upy = my + 3.f*sy, upz = mz + 3.f*sz;

    bool valid = (upx > VMINX) && (upy > VMINY) && (upz > VMINZ) &&
                 (lox < VMAXX) && (loy < VMAXY) && (loz < VMAXZ);

    int lx = (int)((fminf(fmaxf(lox, VMINX), VMAXX) - VMINX) * 2.0f);
    int ly = (int)((fminf(fmaxf(loy, VMINY), VMAXY) - VMINY) * 2.0f);
    int lz = (int)((fminf(fmaxf(loz, VMINZ), VMAXZ) - VMINZ) * 2.0f);
    int hx = min((int)((fminf(fmaxf(upx, VMINX), VMAXX) - VMINX) * 2.0f), GX-1);
    int hy = min((int)((fminf(fmaxf(upy, VMINY), VMAXY) - VMINY) * 2.0f), GY-1);
    int hz = min((int)((fminf(fmaxf(upz, VMINZ), VMAXZ) - VMINZ) * 2.0f), GZ-1);
    hx = min(lx + 7, hx);  hy = min(ly + 7, hy);  hz = min(lz + 7, hz);
    if (!valid) { hx = hy = hz = -1; op = 0.f; }

    // symmetric 3x3 inverse via adjugate
    float a = c11*c22 - c12*c12;
    float b = c02*c12 - c01*c22;
    float c = c01*c12 - c02*c11;
    float det = c00*a + c01*b + c02*c;
    float id  = 1.0f / det;

    GB o;
    o.mx = mx; o.my = my; o.mz = mz; o.op = op;
    o.i00 = a * id;
    o.i01 = 2.f * b * id;
    o.i02 = 2.f * c * id;
    o.i11 = (c00*c22 - c02*c02) * id;
    o.i12 = 2.f * (c02*c01 - c00*c12) * id;
    o.i22 = (c00*c11 - c01*c01) * id;
    o.lx = lx; o.ly = ly; o.lz = lz;
    o.hx = hx; o.hy = hy; o.hz = hz;
    gb[i] = o;

    PK p;
    p.lo = (lx + 1) | ((ly + 1) << 10) | ((lz + 1) << 20);
    p.hi = (hx + 1) | ((hy + 1) << 10) | ((hz + 1) << 20);
    pk[i] = p;
}

// ---------------------------------------------------------------------------
// Kernel 2: one workgroup owns an 8x8x8 voxel tile.
//   A) compact overlapping gaussians (packed-bbox scan, LDS append)
//   B) async-stage batch params+features into LDS, then
//      feats_tile = W(512xG) x F(Gx32) via fp32 WMMA (registers only)
//   C) fused normalization + single streaming store. No global atomics.
// ---------------------------------------------------------------------------
__global__ __launch_bounds__(256)
void splat_kernel(const GB* __restrict__ gb,
                  const PK* __restrict__ pk,
                  const float* __restrict__ feat,
                  float* __restrict__ out)
{
    __shared__ unsigned short list[NG];                 // 16 KB
    __shared__ __align__(16) float sP[BATCH * 16];      // 32 KB staged params
    __shared__ __align__(16) float sF[BATCH * NF];      // 64 KB staged features
    __shared__ int cnt;

    const int tid = threadIdx.x;
    const int bx  = blockIdx.x;          // 25 * 25 * 2 tiles
    const int zt  = bx & 1;
    const int yt  = (bx >> 1) % 25;
    const int xt  = bx / 50;
    const int x0  = xt * 8, y0 = yt * 8, z0 = zt * 8;

    if (tid == 0) cnt = 0;
    __syncthreads();

    // ---- phase A: compact gaussians overlapping this tile ----
    {
        const int px0 = x0 + 1, px1 = x0 + 8;   // biased-coordinate tile bounds
        const int py0 = y0 + 1, py1 = y0 + 8;
        const int pz0 = z0 + 1, pz1 = z0 + 8;
        #pragma unroll 4
        for (int j = 0; j < NG / 256; ++j) {
            int i = j * 256 + tid;
            PK p = pk[i];
            int lx = p.lo & 1023, ly = (p.lo >> 10) & 1023, lz = (p.lo >> 20) & 1023;
            int hx = p.hi & 1023, hy = (p.hi >> 10) & 1023, hz = (p.hi >> 20) & 1023;
            bool ov = (hx >= px0) && (lx <= px1) &&
                      (hy >= py0) && (ly <= py1) &&
                      (hz >= pz0) && (lz <= pz1);
            if (ov) {
                int q = atomicAdd(&cnt, 1);
                list[q] = (unsigned short)i;
            }
        }
    }
    __syncthreads();
    if (tid == 0) {                       // pad to multiple of 4 (WMMA K)
        int c = cnt;
        while (c & 3) list[c++] = (unsigned short)SENT;
        cnt = c;
    }
    __syncthreads();
    const int n = cnt;

    const int lane = tid & 31;
    const int wave = tid >> 5;            // 8 wave32s
    const int half = lane >> 4;           // half-wave: k-pair selector
    const int l16  = lane & 15;

    // per-wave: 4 voxel groups of 16 (M dim), channels split lo/hi (N dim)
    v8f accLo[4], accHi[4];
    float dens[4];
    #pragma unroll
    for (int g = 0; g < 4; ++g) {
        v8f z = {0.f,0.f,0.f,0.f,0.f,0.f,0.f,0.f};
        accLo[g] = z; accHi[g] = z; dens[g] = 0.f;
    }

    // lane's voxel per group (A-matrix: lane L carries row M = L%16)
    float cxv[4], cyv[4], czv[4];
    int   ixv[4], iyv[4], izv[4];
    #pragma unroll
    for (int g = 0; g < 4; ++g) {
        int v  = (wave * 4 + g) * 16 + l16;
        int dz = v & 7, dy = (v >> 3) & 7, dx = v >> 6;
        ixv[g] = x0 + dx; iyv[g] = y0 + dy; izv[g] = z0 + dz;
        cxv[g] = (ixv[g] + 0.5f) * VOXSZ + VMINX;
        cyv[g] = (iyv[g] + 0.5f) * VOXSZ + VMINY;
        czv[g] = (izv[g] + 0.5f) * VOXSZ + VMINZ;
    }

    // ---- phase B: batched LDS staging + WMMA accumulation ----
    for (int s = 0; s < n; s += BATCH) {
        const int m = min(BATCH, n - s);

        // stage params: 4 x 16B per gaussian (async global->LDS)
        for (int c = tid; c < m * 4; c += 256) {
            int g = c >> 2, q = c & 3;
            unsigned gi = list[s + g];
            float* dst = &sP[g * 16 + q * 4];
            if (gi == SENT) {
                uint4 z = {0u, 0u, 0u, 0u};
                *(uint4*)dst = z;                       // op=0 -> zero A column
            } else {
                copy16_g2l((const float*)(gb + gi) + q * 4, dst);
            }
        }
        // stage feature rows: 8 x 16B per gaussian
        for (int c = tid; c < m * 8; c += 256) {
            int g = c >> 3, q = c & 7;
            unsigned gi = list[s + g];
            if (gi == SENT) gi = 0;                     // dead row (A col is zero)
            copy16_g2l(feat + gi * NF + q * 4, &sF[g * NF + q * 4]);
        }
#if USE_ASYNC
        __builtin_amdgcn_s_wait_asynccnt(0);
#endif
        __syncthreads();

        for (int c4 = 0; c4 < m; c4 += 4) {
            // A layout: lanes 0-15 -> K=0,1 ; lanes 16-31 -> K=2,3
            const int slotA = c4 + 2 * half;
            const int slotB = slotA + 1;
            GB pa = *(const GB*)&sP[slotA * 16];
            GB pb = *(const GB*)&sP[slotB * 16];

            // B layout (4x16 fp32): VGPR0 = row kbase, VGPR1 = row kbase+1, N = l16
            v2f bLo, bHi;
            bLo.x = sF[slotA * NF + l16];
            bLo.y = sF[slotB * NF + l16];
            bHi.x = sF[slotA * NF + 16 + l16];
            bHi.y = sF[slotB * NF + 16 + l16];

            #pragma unroll
            for (int g = 0; g < 4; ++g) {
                // branchless weight: exp evaluated unconditionally, in-box test
                // folded in as a 0/1 multiplier (keeps EXEC full, single block)
                float dxa = cxv[g] - pa.mx, dya = cyv[g] - pa.my, dza = czv[g] - pa.mz;
                float mska = ((ixv[g] >= pa.lx) & (ixv[g] <= pa.hx) &
                              (iyv[g] >= pa.ly) & (iyv[g] <= pa.hy) &
                              (izv[g] >= pa.lz) & (izv[g] <= pa.hz)) ? 1.0f : 0.0f;
                float ma = pa.i00*dxa*dxa + pa.i11*dya*dya + pa.i22*dza*dza
                         + pa.i01*dxa*dya + pa.i02*dxa*dza + pa.i12*dya*dza;
                float wa = pa.op * mska * __expf(-0.5f * ma);

                float dxb = cxv[g] - pb.mx, dyb = cyv[g] - pb.my, dzb = czv[g] - pb.mz;
                float mskb = ((ixv[g] >= pb.lx) & (ixv[g] <= pb.hx) &
                              (iyv[g] >= pb.ly) & (iyv[g] <= pb.hy) &
                              (izv[g] >= pb.lz) & (izv[g] <= pb.hz)) ? 1.0f : 0.0f;
                float mb = pb.i00*dxb*dxb + pb.i11*dyb*dyb + pb.i22*dzb*dzb
                         + pb.i01*dxb*dyb + pb.i02*dxb*dzb + pb.i12*dyb*dzb;
                float wb = pb.op * mskb * __expf(-0.5f * mb);

                v2f av; av.x = wa; av.y = wb;
                dens[g] += wa + wb;

                // D(16x16) += A(16x4) x B(4x16), full fp32
                accLo[g] = __builtin_amdgcn_wmma_f32_16x16x4_f32(
                    false, av, false, bLo, (short)0, accLo[g], false, false);
                accHi[g] = __builtin_amdgcn_wmma_f32_16x16x4_f32(
                    false, av, false, bHi, (short)0, accHi[g], false, false);
            }
        }
        __syncthreads();
    }

    // ---- phase C: normalize + single streaming store ----
    float* fo = out + (size_t)GX * GY * GZ;   // feats follow density
    #pragma unroll
    for (int g = 0; g < 4; ++g) {
        float d = dens[g] + __shfl_xor(dens[g], 16, 32);   // sum both k-halves
        int grp = wave * 4 + g;

        if (half == 0) {                                    // density, 1 lane/voxel
            int v  = grp * 16 + l16;
            int dz = v & 7, dy = (v >> 3) & 7, dx = v >> 6;
            int flat = ((x0 + dx) * GY + (y0 + dy)) * GZ + (z0 + dz);
            out[flat] = d;
        }

        // D layout: VGPR r -> voxel M = r + 8*half, channel N = l16 (+16 for Hi)
        #pragma unroll
        for (int r = 0; r < 8; ++r) {
            int   M   = r + 8 * half;
            float dm  = __shfl(d, M, 32);                   // density of voxel M
            float inv = 1.0f / fmaxf(dm, EPSV);
            int v  = grp * 16 + M;
            int dz = v & 7, dy = (v >> 3) & 7, dx = v >> 6;
            int flat = ((x0 + dx) * GY + (y0 + dy)) * GZ + (z0 + dz);
            fo[(size_t)flat * NF + l16]      = accLo[g][r] * inv;
            fo[(size_t)flat * NF + 16 + l16] = accHi[g][r] * inv;
        }
    }
}

// ---------------------------------------------------------------------------
extern "C" void kernel_launch(void* const* d_in, const int* in_sizes, int n_in,
                              void* d_out, int out_size, void* d_ws, size_t ws_size,
                              hipStream_t stream) {
    const float* means = (const float*)d_in[0];   // (8192,3)
    const float* ops   = (const float*)d_in[1];   // (8192,1)
    const float* cov   = (const float*)d_in[2];   // (8192,3,3)
    const float* feat  = (const float*)d_in[3];   // (8192,32)

    GB* gb = (GB*)d_ws;
    PK* pk = (PK*)((char*)d_ws + (size_t)NG * sizeof(GB));

    prep_kernel<<<NG / 256, 256, 0, stream>>>(means, ops, cov, gb, pk);
    splat_kernel<<<25 * 25 * 2, 256, 0, stream>>>(gb, pk, feat, (float*)d_out);
}